// CompactBilinearPooling_40235253629205
// MI455X (gfx1250) — compile-verified
//
#include <hip/hip_runtime.h>
#include <math.h>

#define D_OUT 16000
#define D_IN  2048
#define BATCH 512
#define NTHR_FFT 512   // 16 wave32s per workgroup

typedef float v2f __attribute__((ext_vector_type(2)));
typedef float v8f __attribute__((ext_vector_type(8)));

__device__ __forceinline__ float2 cadd(float2 a, float2 b) { return make_float2(a.x + b.x, a.y + b.y); }
__device__ __forceinline__ float2 csub(float2 a, float2 b) { return make_float2(a.x - b.x, a.y - b.y); }
__device__ __forceinline__ float2 cmul(float2 a, float2 b) {
  return make_float2(a.x * b.x - a.y * b.y, a.x * b.y + a.y * b.x);
}

// ---------------------------------------------------------------------------
// All FFT stages address the dynamic-LDS symbol directly with integer offsets
// (0 or D_OUT) so every access is statically addrspace(3) -> ds_load/ds_store,
// not flat_load (which would tie up both DScnt and LOADcnt on CDNA5).
// ---------------------------------------------------------------------------

// Stage 1 of the length-16000 Stockham FFT: radix-16 DIF, s=1.
//   y[16p + u] = ( sum_t W16[u,t] * x[p + 1000 t] ) * w16000^(u*p)
// DFT-16 done as V_WMMA_F32_16X16X4_F32: A = W16 rows resident per-lane
// (A 16x4 f32 layout: lanes 0-15 = M, VGPR0/1 = K, lanes 16-31 = K+2),
// B/N dimension = 16 butterfly columns, complex = 4 real WMMAs per K-step.
__device__ void radix16_stage(int inOff, int outOff, float sign, int tid) {
  extern __shared__ float2 smem[];
  const int lane = tid & 31;
  const int wave = tid >> 5;
  const int m    = lane & 15;   // A row (u), B/C column-in-tile
  const int kg   = lane >> 4;   // 0/1: K sub-group per ISA layout

  // Per-lane row m of the DFT-16 matrix: w[t] = exp(sign*i*2pi*m*t/16)
  float wre[16], wim[16];
#pragma unroll
  for (int t = 0; t < 16; ++t) {
    float ang = sign * (6.2831853071795864f / 16.0f) * (float)(m * t);
    __sincosf(ang, &wim[t], &wre[t]);
  }

  for (int tile = wave; tile < 63; tile += NTHR_FFT / 32) {  // ceil(1000/16)=63
    const int p0   = tile * 16;
    const int pcol = p0 + m;
    const int pl   = pcol < 1000 ? pcol : 999;  // clamp tail loads (column-local garbage, masked at store)

    v8f cre = {0.f, 0.f, 0.f, 0.f, 0.f, 0.f, 0.f, 0.f};
    v8f cim = {0.f, 0.f, 0.f, 0.f, 0.f, 0.f, 0.f, 0.f};

#pragma unroll
    for (int kk = 0; kk < 16; kk += 4) {
      const int ka = kk + kg * 2;
      v2f are  = {  wre[ka],  wre[ka + 1] };
      v2f aim  = {  wim[ka],  wim[ka + 1] };
      v2f aimn = { -wim[ka], -wim[ka + 1] };  // f32 WMMA has no A-negate (NEG=CNeg only)
      float2 b0 = smem[inOff + pl + 1000 * ka];
      float2 b1 = smem[inOff + pl + 1000 * (ka + 1)];
      v2f bre = { b0.x, b1.x };
      v2f bim = { b0.y, b1.y };
      // complex C += W * B  as 4 real f32 WMMAs
      cre = __builtin_amdgcn_wmma_f32_16x16x4_f32(false, are,  false, bre, (short)0, cre, false, false);
      cre = __builtin_amdgcn_wmma_f32_16x16x4_f32(false, aimn, false, bim, (short)0, cre, false, false);
      cim = __builtin_amdgcn_wmma_f32_16x16x4_f32(false, are,  false, bim, (short)0, cim, false, false);
      cim = __builtin_amdgcn_wmma_f32_16x16x4_f32(false, aim,  false, bre, (short)0, cim, false, false);
    }

    if (pcol < 1000) {
      // C/D layout: VGPR r -> M = r + 8*kg ; N = lane&15. Apply w16000^(u*p) twiddle, store.
#pragma unroll
      for (int r = 0; r < 8; ++r) {
        const int u = r + kg * 8;
        float ang = sign * (6.2831853071795864f / 16000.0f) * (float)(u * pcol);
        float tr, ti;
        __sincosf(ang, &ti, &tr);
        float yr = cre[r], yi = cim[r];
        smem[outOff + 16 * pcol + u] = make_float2(yr * tr - yi * ti, yr * ti + yi * tr);
      }
    }
  }
}

// Generic Stockham DIF radix-2 stage: n current length, s stride (s*n == 16000)
__device__ void radix2_stage(int inOff, int outOff, int n, int s, float sign, int tid) {
  extern __shared__ float2 smem[];
  const int m = n >> 1;
  const int total = m * s;  // 8000
  for (int idx = tid; idx < total; idx += NTHR_FFT) {
    const int p = idx / s;
    const int q = idx - p * s;
    float2 a = smem[inOff + q + s * p];
    float2 b = smem[inOff + q + s * (p + m)];
    float ang = sign * 6.2831853071795864f * (float)p / (float)n;
    float wr, wi;
    __sincosf(ang, &wi, &wr);
    float2 d = csub(a, b);
    smem[outOff + q + s * (2 * p)]     = cadd(a, b);
    smem[outOff + q + s * (2 * p + 1)] = make_float2(d.x * wr - d.y * wi, d.x * wi + d.y * wr);
  }
}

// Generic Stockham DIF radix-5 stage
__device__ void radix5_stage(int inOff, int outOff, int n, int s, float sign, int tid) {
  extern __shared__ float2 smem[];
  const int m = n / 5;
  const int total = m * s;  // 3200
  float2 w5[5];
  w5[0] = make_float2(1.f, 0.f);
  {
    float ang5 = sign * 6.2831853071795864f / 5.0f;
    __sincosf(ang5, &w5[1].y, &w5[1].x);
    w5[2] = cmul(w5[1], w5[1]);
    w5[3] = cmul(w5[2], w5[1]);
    w5[4] = cmul(w5[3], w5[1]);
  }
  for (int idx = tid; idx < total; idx += NTHR_FFT) {
    const int p = idx / s;
    const int q = idx - p * s;
    float2 a[5];
#pragma unroll
    for (int t = 0; t < 5; ++t) a[t] = smem[inOff + q + s * (p + t * m)];
#pragma unroll
    for (int u = 0; u < 5; ++u) {
      float2 acc = a[0];
#pragma unroll
      for (int t = 1; t < 5; ++t) acc = cadd(acc, cmul(a[t], w5[(u * t) % 5]));
      float ang = sign * 6.2831853071795864f * (float)(u * p) / (float)n;
      float wr, wi;
      __sincosf(ang, &wi, &wr);
      smem[outOff + q + s * (5 * p + u)] = make_float2(acc.x * wr - acc.y * wi, acc.x * wi + acc.y * wr);
    }
  }
}

// Full length-16000 Stockham FFT (sign=-1 fwd, +1 unnormalized inverse).
// Plan: 16 -> 2,2,2 -> 5,5,5. Ping-pongs LDS *offsets*; returns result offset.
__device__ int fft16000(int xOff, int yOff, float sign, int tid) {
  radix16_stage(xOff, yOff, sign, tid);
  __syncthreads();
  int cur = yOff, oth = xOff;
  int n = 1000, s = 16;
  for (int st = 0; st < 3; ++st) {
    radix2_stage(cur, oth, n, s, sign, tid);
    __syncthreads();
    int t = cur; cur = oth; oth = t;
    n >>= 1; s <<= 1;
  }
  for (int st = 0; st < 3; ++st) {
    radix5_stage(cur, oth, n, s, sign, tid);
    __syncthreads();
    int t = cur; cur = oth; oth = t;
    n /= 5; s *= 5;
  }
  return cur;
}

// ---------------------------------------------------------------------------
// K1: recover (h, s) of each count-sketch row while doing the mandatory 262MB scan
// ---------------------------------------------------------------------------
__global__ void extract_sketch_kernel(const float* __restrict__ sk1,
                                      const float* __restrict__ sk2,
                                      int* __restrict__ h1, float* __restrict__ s1,
                                      int* __restrict__ h2, float* __restrict__ s2) {
  const float* sk = (blockIdx.y == 0) ? sk1 : sk2;
  int*   h = (blockIdx.y == 0) ? h1 : h2;
  float* s = (blockIdx.y == 0) ? s1 : s2;
  const int row = blockIdx.x;
  const float4* r4 = reinterpret_cast<const float4*>(sk + (size_t)row * D_OUT);
  for (int j = threadIdx.x; j < D_OUT / 4; j += blockDim.x) {
    float4 v = r4[j];
    if (v.x != 0.f) { h[row] = 4 * j + 0; s[row] = v.x; }
    if (v.y != 0.f) { h[row] = 4 * j + 1; s[row] = v.y; }
    if (v.z != 0.f) { h[row] = 4 * j + 2; s[row] = v.z; }
    if (v.w != 0.f) { h[row] = 4 * j + 3; s[row] = v.w; }
  }
}

// ---------------------------------------------------------------------------
// K2: one workgroup per batch row. Sparse projection scatter into LDS
// (z = p1 + i*p2) via ds_add_f32, fwd FFT, Hermitian split + product,
// inverse FFT, signed sqrt. LDS: 2 x 16000 complex = 256000 B (CDNA5 320KB WGP LDS).
// ---------------------------------------------------------------------------
__global__ __launch_bounds__(NTHR_FFT)
void cbp_fft_kernel(const float* __restrict__ x1, const float* __restrict__ x2,
                    const int* __restrict__ h1, const float* __restrict__ s1,
                    const int* __restrict__ h2, const float* __restrict__ s2,
                    float* __restrict__ out) {
  extern __shared__ float2 smem[];
  const int tid = threadIdx.x;
  const int b   = blockIdx.x;

  for (int i = tid; i < D_OUT; i += NTHR_FFT) smem[i] = make_float2(0.f, 0.f);
  __syncthreads();

  // count-sketch projection == the dense GEMM row, in 2048 LDS MACs
  const float* x1r = x1 + (size_t)b * D_IN;
  const float* x2r = x2 + (size_t)b * D_IN;
  for (int i = tid; i < D_IN; i += NTHR_FFT) {
    atomicAdd(&smem[h1[i]].x, s1[i] * x1r[i]);
    atomicAdd(&smem[h2[i]].y, s2[i] * x2r[i]);
  }
  __syncthreads();

  const int zOff = fft16000(0, D_OUT, -1.0f, tid);

  // Two-for-one split: F1=(Z[k]+conj(Z[N-k]))/2, F2=(Z[k]-conj(Z[N-k]))/(2i); G=F1*F2.
  // Each k in [0,8000] owns both k and N-k -> no cross-thread conflicts.
  for (int k = tid; k <= D_OUT / 2; k += NTHR_FFT) {
    const int k2 = (k == 0) ? 0 : (D_OUT - k);
    float2 z1 = smem[zOff + k];
    float2 z2 = smem[zOff + k2];
    float2 f1 = make_float2(0.5f * (z1.x + z2.x), 0.5f * (z1.y - z2.y));
    float2 f2 = make_float2(0.5f * (z1.y + z2.y), 0.5f * (z2.x - z1.x));
    float2 g  = cmul(f1, f2);
    smem[zOff + k] = g;
    if (k2 != k) smem[zOff + k2] = make_float2(g.x, -g.y);  // Hermitian
  }
  __syncthreads();

  // Unnormalized inverse == reference's irfft(...)*d (the factors cancel)
  const int rOff = fft16000(zOff, zOff == 0 ? D_OUT : 0, +1.0f, tid);

  float* orow = out + (size_t)b * D_OUT;
  for (int i = tid; i < D_OUT; i += NTHR_FFT) {
    float c = smem[rOff + i].x;              // conj-symmetric spectrum -> real
    orow[i] = copysignf(sqrtf(fabsf(c)), c); // signed sqrt
  }
}

// K3: column (batch-dim) sum of squares; coalesced across k
__global__ void colsq_kernel(const float* __restrict__ out, float* __restrict__ sq) {
  const int k = blockIdx.x * blockDim.x + threadIdx.x;
  if (k >= D_OUT) return;
  float acc = 0.f;
  for (int b = 0; b < BATCH; ++b) {
    float v = out[(size_t)b * D_OUT + k];
    acc += v * v;
  }
  sq[k] = acc;
}

// K4: L2 normalize along batch dim
__global__ void norm_kernel(float* __restrict__ out, const float* __restrict__ sq) {
  const size_t idx = (size_t)blockIdx.x * blockDim.x + threadIdx.x;
  if (idx >= (size_t)BATCH * D_OUT) return;
  const int k = (int)(idx % D_OUT);
  const float n = sqrtf(sq[k]);
  out[idx] = out[idx] / fmaxf(n, 1e-12f);
}

extern "C" void kernel_launch(void* const* d_in, const int* in_sizes, int n_in,
                              void* d_out, int out_size, void* d_ws, size_t ws_size,
                              hipStream_t stream) {
  const float* x1  = (const float*)d_in[0];
  const float* x2  = (const float*)d_in[1];
  const float* sk1 = (const float*)d_in[2];
  const float* sk2 = (const float*)d_in[3];
  float* out = (float*)d_out;

  char*  ws = (char*)d_ws;
  int*   h1 = (int*)(ws + 0);
  float* s1 = (float*)(ws + 8192);
  int*   h2 = (int*)(ws + 16384);
  float* s2 = (float*)(ws + 24576);
  float* sq = (float*)(ws + 32768);   // 64KB; total ws use < 128KB

  extract_sketch_kernel<<<dim3(D_IN, 2), 256, 0, stream>>>(sk1, sk2, h1, s1, h2, s2);

  const size_t lds_bytes = 2 * D_OUT * sizeof(float2);  // 256000 B
  cbp_fft_kernel<<<BATCH, NTHR_FFT, lds_bytes, stream>>>(x1, x2, h1, s1, h2, s2, out);

  colsq_kernel<<<(D_OUT + 255) / 256, 256, 0, stream>>>(out, sq);
  norm_kernel<<<((size_t)BATCH * D_OUT + 255) / 256, 256, 0, stream>>>(out, sq);
}